// QNetwork_13125420057138
// MI455X (gfx1250) — compile-verified
//
#include <hip/hip_runtime.h>

#define NN   100000
#define EE   3200000
#define NGNB 50000
#define FINK 6
#define KPAD 8
#define HDIM 64
#define FOUT 5

typedef __attribute__((ext_vector_type(2))) float v2f;
typedef __attribute__((ext_vector_type(8))) float v8f;

// ---------------- degree / normalization ----------------
__global__ void deg_init(float* __restrict__ deg) {
    int i = blockIdx.x * blockDim.x + threadIdx.x;
    if (i < NN) deg[i] = 1.0f;                      // self-loop weight
}

__global__ void deg_accum(const int* __restrict__ dst, const float* __restrict__ w,
                          float* __restrict__ deg) {
    int e = blockIdx.x * blockDim.x + threadIdx.x;
    if (e < EE) atomicAdd(&deg[dst[e]], w[e]);
}

__global__ void make_dinv(const float* __restrict__ deg, float* __restrict__ dinv) {
    int i = blockIdx.x * blockDim.x + threadIdx.x;
    if (i < NN) {
        float d = deg[i];
        dinv[i] = (d > 0.0f) ? rsqrtf(d) : 0.0f;
    }
}

// ---------------- input padding: x0[N,8] = vf[N,6] zero-padded ----------------
__global__ void pad_input(const float* __restrict__ vf, float* __restrict__ x0) {
    int i = blockIdx.x * blockDim.x + threadIdx.x;
    if (i < NN * KPAD) {
        int node = i >> 3;
        int c = i & 7;
        x0[i] = (c < FINK) ? vf[node * FINK + c] : 0.0f;
    }
}

// ---------------- dense layer GEMM: Y[N,64] = X[N,K] @ W[K,64] ----------------
// Compile-time K (multiple of 4) -> no bounds guards, no EXEC churn, fully
// unrolled chain of V_WMMA_F32_16X16X4_F32. One wave = 16-row strip x 64 cols.
template <int K>
__global__ void gemm_wmma(const float* __restrict__ X, const float* __restrict__ W,
                          float* __restrict__ Y) {
    int gtid = blockIdx.x * blockDim.x + threadIdx.x;
    int wave = gtid >> 5;
    int lane = threadIdx.x & 31;
    int row0 = wave << 4;                 // 16 rows per wave; NN % 16 == 0
    if (row0 >= NN) return;               // wave-uniform: EXEC stays all-ones

    const int mA   = lane & 15;           // A: M index (both lane halves)
    const int kSub = (lane >> 4) << 1;    // A/B: k sub-offset 0 or 2
    const int nB   = lane & 15;           // B/D: N index within 16-col tile

    v8f acc[4] = {{}, {}, {}, {}};
    // Each lane's k-pair is contiguous & 8B-aligned -> b64 loads.
    const v2f* xr = (const v2f*)(X + (size_t)(row0 + mA) * K);

#pragma unroll
    for (int k0 = 0; k0 < K; k0 += 4) {
        v2f a = xr[(k0 + kSub) >> 1];
#pragma unroll
        for (int t = 0; t < 4; ++t) {
            int col = (t << 4) + nB;
            v2f b;
            b.x = W[(k0 + kSub) * HDIM + col];
            b.y = W[(k0 + kSub + 1) * HDIM + col];
            acc[t] = __builtin_amdgcn_wmma_f32_16x16x4_f32(
                false, a, false, b, (short)0, acc[t], false, false);
        }
    }

    // D layout: lanes 0-15 -> M = r, lanes 16-31 -> M = 8 + r; N = lane & 15
    int mBase = (lane >> 4) << 3;
#pragma unroll
    for (int t = 0; t < 4; ++t) {
#pragma unroll
        for (int r = 0; r < 8; ++r) {
            int row = row0 + mBase + r;
            Y[(size_t)row * HDIM + (t << 4) + nB] = acc[t][r];
        }
    }
}

// ---------------- aggregation ----------------
// agg[i] = y[i] * dinv[i]^2   (self-loop term; also zero-initializes agg)
__global__ void self_loop_init(const float* __restrict__ Y, const float* __restrict__ dinv,
                               float* __restrict__ agg) {
    int i = blockIdx.x * blockDim.x + threadIdx.x;
    if (i < NN * HDIM) {
        float di = dinv[i >> 6];
        agg[i] = Y[i] * di * di;
    }
}

// 16 lanes per edge; each handles a float4 slice of the 64-wide feature row.
__global__ void edge_scatter(const int* __restrict__ src, const int* __restrict__ dst,
                             const float* __restrict__ w, const float* __restrict__ dinv,
                             const float* __restrict__ Y, float* __restrict__ agg) {
    int idx = blockIdx.x * blockDim.x + threadIdx.x;
    int e = idx >> 4;
    if (e >= EE) return;
    int p = idx & 15;
    int s = src[e];
    int d = dst[e];
    float nrm = dinv[s] * w[e] * dinv[d];
    const float4* ys = (const float4*)(Y + ((size_t)s << 6));
    float4 v = ys[p];
    float* o = agg + ((size_t)d << 6) + (p << 2);
    atomicAdd(o + 0, v.x * nrm);
    atomicAdd(o + 1, v.y * nrm);
    atomicAdd(o + 2, v.z * nrm);
    atomicAdd(o + 3, v.w * nrm);
}

__global__ void bias_sigmoid(const float* __restrict__ agg, const float* __restrict__ b,
                             float* __restrict__ X) {
    int i = blockIdx.x * blockDim.x + threadIdx.x;
    if (i < NN * HDIM) {
        float v = agg[i] + b[i & 63];
        X[i] = 1.0f / (1.0f + __expf(-v));
    }
}

// ---------------- head: out[i,c] = x[i,:] @ Wl[:,c] + bl[c], i < 50000 ----------------
__global__ void head(const float* __restrict__ X, const float* __restrict__ Wl,
                     const float* __restrict__ bl, float* __restrict__ out) {
    int idx = blockIdx.x * blockDim.x + threadIdx.x;
    if (idx >= NGNB * FOUT) return;
    int i = idx / FOUT;
    int c = idx % FOUT;
    const float* xr = X + (size_t)i * HDIM;
    float acc = bl[c];
#pragma unroll
    for (int j = 0; j < HDIM; ++j) acc = fmaf(xr[j], Wl[j * FOUT + c], acc);
    out[idx] = acc;
}

extern "C" void kernel_launch(void* const* d_in, const int* in_sizes, int n_in,
                              void* d_out, int out_size, void* d_ws, size_t ws_size,
                              hipStream_t stream) {
    (void)in_sizes; (void)n_in; (void)out_size; (void)ws_size;

    const float* vf  = (const float*)d_in[0];   // [N, 6]
    const int*   edg = (const int*)  d_in[1];   // [2, E]
    const float* wgt = (const float*)d_in[2];   // [E]
    const float* Ws[4] = {(const float*)d_in[3], (const float*)d_in[5],
                          (const float*)d_in[7], (const float*)d_in[9]};
    const float* Bs[4] = {(const float*)d_in[4], (const float*)d_in[6],
                          (const float*)d_in[8], (const float*)d_in[10]};
    const float* Wl = (const float*)d_in[11];
    const float* bl = (const float*)d_in[12];
    const int* src = edg;
    const int* dst = edg + EE;

    // workspace (fp32): deg[N] | dinv[N] | x0[N*8] | x[N*64] | y[N*64] | agg[N*64]
    float* ws   = (float*)d_ws;
    float* deg  = ws;
    float* dinv = ws + NN;
    float* x0   = ws + 2 * NN;
    float* xb   = x0 + (size_t)NN * KPAD;
    float* yb   = xb + (size_t)NN * HDIM;
    float* ag   = yb + (size_t)NN * HDIM;

    const int B = 256;
    deg_init <<<(NN + B - 1) / B, B, 0, stream>>>(deg);
    deg_accum<<<(EE + B - 1) / B, B, 0, stream>>>(dst, wgt, deg);
    make_dinv<<<(NN + B - 1) / B, B, 0, stream>>>(deg, dinv);
    pad_input<<<(NN * KPAD + B - 1) / B, B, 0, stream>>>(vf, x0);

    const int waves      = NN / 16;                    // 6250 strips
    const int gemmBlocks = (waves * 32 + 127) / 128;   // 4 waves / block
    const int nh         = NN * HDIM;
    const int scatterWrk = EE * 16;

    for (int l = 0; l < 4; ++l) {
        if (l == 0)
            gemm_wmma<KPAD><<<gemmBlocks, 128, 0, stream>>>(x0, Ws[0], yb);
        else
            gemm_wmma<HDIM><<<gemmBlocks, 128, 0, stream>>>(xb, Ws[l], yb);
        self_loop_init<<<(nh + B - 1) / B, B, 0, stream>>>(yb, dinv, ag);
        edge_scatter  <<<(scatterWrk + B - 1) / B, B, 0, stream>>>(src, dst, wgt, dinv, yb, ag);
        bias_sigmoid  <<<(nh + B - 1) / B, B, 0, stream>>>(ag, Bs[l], xb);
    }

    head<<<(NGNB * FOUT + B - 1) / B, B, 0, stream>>>(xb, Wl, bl, (float*)d_out);
}